// HausdorffDistanceLoss_16329465659786
// MI455X (gfx1250) — compile-verified
//
#include <hip/hip_runtime.h>
#include <stdint.h>

#define HD_MAXD   20
#define HD_H      1024
#define HD_W      1024
#define HD_WW     32                       // 32-bit words per packed row
#define HD_CORE   128                      // core rows per strip
#define HD_EXT    (HD_CORE + 2 * HD_MAXD)  // 168 rows incl. halos
#define HD_STRIPS (HD_H / HD_CORE)         // 8 strips per image
#define HD_ONES   0xFFFFFFFFu

// ---------------------------------------------------------------------------
// Zero the 4 u32 accumulators (workspace is poisoned before timing).
__global__ void hd_zero_acc(uint32_t* __restrict__ acc) {
  if (threadIdx.x < 4) acc[threadIdx.x] = 0u;
}

// ---------------------------------------------------------------------------
// Binarize + bit-pack both masks with wave32 ballot.
//   mp bit = 1  <=>  pred <= 0   (mask for pred distance transform)
//   mt bit = 1  <=>  target == 0 (mask for target distance transform)
__global__ void hd_pack(const float* __restrict__ pred,
                        const int* __restrict__ target,
                        uint32_t* __restrict__ mp,
                        uint32_t* __restrict__ mt) {
  int g = blockIdx.x * blockDim.x + threadIdx.x;
  float p = pred[g];
  int t = target[g];
  uint32_t bp = __builtin_amdgcn_ballot_w32(p > 0.0f);  // pred_binary bits
  uint32_t bt = __builtin_amdgcn_ballot_w32(t != 0);    // target==1 bits
  if ((threadIdx.x & 31) == 0) {
    mp[g >> 5] = ~bp;
    mt[g >> 5] = ~bt;
  }
}

// ---------------------------------------------------------------------------
// 20 rounds of 3x3 binary erosion on a packed strip held in LDS.
// blockIdx: [img(4b)][which(1b)][strip(3b)]
//   which==0: erode pred-mask, weight = target==1 bits  -> acc[0], acc[2]
//   which==1: erode target-mask, weight = pred>0 bits   -> acc[1], acc[3]
__global__ __launch_bounds__(1024) void hd_erode(const uint32_t* __restrict__ mp,
                                                 const uint32_t* __restrict__ mt,
                                                 uint32_t* __restrict__ acc) {
  extern __shared__ uint32_t smem[];
  uint32_t* bufA = smem;                  // HD_EXT * 32 words
  uint32_t* bufB = smem + HD_EXT * HD_WW; // ping-pong buffer

  const int blk   = blockIdx.x;
  const int strip = blk & (HD_STRIPS - 1);
  const int which = (blk >> 3) & 1;
  const int img   = blk >> 4;

  const uint32_t* __restrict__ srcA = which ? mt : mp;
  const uint32_t* __restrict__ srcW = which ? mp : mt;

  const int tid = threadIdx.x;
  const int wc  = tid & 31;    // word column (lane)
  const int rb  = tid >> 5;    // row group (wave id)
  const int e0  = rb * 6;      // first ext row owned by this thread (28 active groups)
  const int s0  = strip * HD_CORE;
  const size_t ibase = (size_t)img * HD_H * HD_WW;

  // Weight words (complement of the *other* mask) for this thread's core rows.
  uint32_t wgt[6];
  uint32_t accw = 0;
  #pragma unroll
  for (int i = 0; i < 6; ++i) {
    int e = e0 + i;
    uint32_t w = 0u;
    if (e >= HD_MAXD && e < HD_MAXD + HD_CORE) {
      int rg = s0 + (e - HD_MAXD);
      w = ~srcW[ibase + (size_t)rg * HD_WW + wc];
    }
    wgt[i] = w;
    accw += __popc(w);
  }

  // Stage packed strip (rows s0-20 .. s0+147) into LDS bufA.
  bool interior = (s0 >= HD_MAXD) && (s0 + HD_CORE + HD_MAXD <= HD_H);
  if (interior) {
    // gfx1250 async copy: 1344 x 16B chunks, LDS layout == global layout.
    uint32_t lbase = (uint32_t)(size_t)(void*)bufA;
    uint64_t gbase = (uint64_t)(size_t)(const void*)(srcA + ibase +
                                                     (size_t)(s0 - HD_MAXD) * HD_WW);
    #pragma unroll
    for (int j = 0; j < 2; ++j) {
      int c = tid + j * 1024;
      if (c < (HD_EXT * HD_WW * 4) / 16) {
        uint32_t lo = lbase + (uint32_t)c * 16u;
        uint64_t ga = gbase + (uint64_t)c * 16u;
        asm volatile("global_load_async_to_lds_b128 %0, %1, off"
                     :: "v"(lo), "v"(ga) : "memory");
      }
    }
    asm volatile("s_wait_asynccnt 0" ::: "memory");
  } else {
    // Border strips: clamp out-of-image rows to all-ones (inf padding == no erosion).
    for (int e = rb; e < HD_EXT; e += 32) {
      int rg = s0 - HD_MAXD + e;
      uint32_t v = HD_ONES;
      if (rg >= 0 && rg < HD_H) v = srcA[ibase + (size_t)rg * HD_WW + wc];
      bufA[e * HD_WW + wc] = v;
    }
  }
  __syncthreads();

  // Sum_{d=0..19} popc(after_d & weight)  ==  sum(dist_clamped * weight).
  uint32_t accv = 0;
  uint32_t* A  = bufA;
  uint32_t* Bp = bufB;
  const int lanex4 = wc << 2;
  for (int d = 0; d < HD_MAXD; ++d) {
    if (e0 < HD_EXT) {  // row groups 0..27 active (uniform per wave)
      uint32_t wprev = (e0 == 0) ? HD_ONES : A[(e0 - 1) * HD_WW + wc];
      uint32_t wcur  = A[e0 * HD_WW + wc];
      #pragma unroll
      for (int i = 0; i < 6; ++i) {
        int e = e0 + i;
        uint32_t wnext = (e == HD_EXT - 1) ? HD_ONES : A[(e + 1) * HD_WW + wc];
        accv += __popc(wcur & wgt[i]);
        // vertical AND then horizontal AND (1-bit shifts, carries via bpermute)
        uint32_t V  = wprev & wcur & wnext;
        uint32_t vp = (uint32_t)__builtin_amdgcn_ds_bpermute(lanex4 - 4, (int)V);
        uint32_t vn = (uint32_t)__builtin_amdgcn_ds_bpermute(lanex4 + 4, (int)V);
        if (wc == 0)  vp = HD_ONES;   // word column -1: outside image -> ones
        if (wc == 31) vn = HD_ONES;   // word column 32: outside image -> ones
        uint32_t right = (V >> 1) | ((vn & 1u) << 31);
        uint32_t left  = (V << 1) | (vp >> 31);
        Bp[e * HD_WW + wc] = V & right & left;
        wprev = wcur;
        wcur  = wnext;
      }
    }
    __syncthreads();
    uint32_t* t = A; A = Bp; Bp = t;
  }

  atomicAdd(&acc[which], accv);      // S1 or S2 numerator sum
  atomicAdd(&acc[2 + which], accw);  // sum(t) or sum(pred_binary)
}

// ---------------------------------------------------------------------------
__global__ void hd_finalize(const uint32_t* __restrict__ acc, float* __restrict__ out) {
  float S1 = (float)acc[0];
  float S2 = (float)acc[1];
  float st = (float)acc[2];
  float sp = (float)acc[3];
  float hf = (S1 + 1e-6f) / (st + 1e-6f);
  float hb = (S2 + 1e-6f) / (sp + 1e-6f);
  out[0] = 0.5f * (hf + hb);
}

// ---------------------------------------------------------------------------
extern "C" void kernel_launch(void* const* d_in, const int* in_sizes, int n_in,
                              void* d_out, int out_size, void* d_ws, size_t ws_size,
                              hipStream_t stream) {
  (void)n_in; (void)out_size; (void)ws_size;
  const float* pred   = (const float*)d_in[0];
  const int*   target = (const int*)d_in[1];

  const int n      = in_sizes[0];            // 16 * 1024 * 1024
  const int nimg   = n / (HD_H * HD_W);      // 16
  const int nwords = n / 32;                 // packed words per mask

  uint32_t* acc = (uint32_t*)d_ws;                     // 4 u32 accumulators
  uint32_t* mp  = (uint32_t*)((char*)d_ws + 256);      // packed pred mask (2 MB)
  uint32_t* mt  = mp + nwords;                         // packed target mask (2 MB)

  hipLaunchKernelGGL(hd_zero_acc, dim3(1), dim3(32), 0, stream, acc);
  hipLaunchKernelGGL(hd_pack, dim3(n / 256), dim3(256), 0, stream,
                     pred, target, mp, mt);

  const int    nblk  = nimg * 2 * HD_STRIPS;                 // 256 blocks
  const size_t shmem = (size_t)HD_EXT * HD_WW * 4 * 2;       // 43 KB ping-pong
  hipLaunchKernelGGL(hd_erode, dim3(nblk), dim3(1024), shmem, stream, mp, mt, acc);

  hipLaunchKernelGGL(hd_finalize, dim3(1), dim3(1), 0, stream, acc, (float*)d_out);
}